// W2DMam_31190052503711
// MI455X (gfx1250) — compile-verified
//
#include <hip/hip_runtime.h>

#define DM   64
#define DI   128
#define NDS  16
#define DR   4
#define L    64

typedef __attribute__((ext_vector_type(2))) float v2f;
typedef __attribute__((ext_vector_type(8))) float v8f;

// ---- LDS layout (floats), even row strides for b64/b128 LDS ops ----
// [0      .. 4224)  x window [l][m] (64x66)   -> reused: xdblT [l][c] (64x48)
// [4224   .. 12672) xc  [c][l] (128x66)       -> reused: ymerge [d][pos]
// [12672  .. 20992) z   [l][c] (64x130)
// [20992  .. 29440) xcv [c][l] (128x66)  (post conv+silu)
// [29440  .. 37888) dt  [d][l] (128x66)       -> reused: ya [l][d] (64x130)
#define S_XLM  0
#define S_XDBL 0
#define S_XC   4224
#define S_YM   4224
#define S_Z    12672
#define S_XCV  20992
#define S_DT   29440
#define S_YA   29440
#define SMEM_FLOATS 37888

__device__ __forceinline__ int posmap(int k, int t) {
    // involutive map: same for input gather and output merge
    if (k == 0) return t;
    if (k == 1) return ((t & 7) << 3) | (t >> 3);
    if (k == 2) return 63 - t;
    int l = 63 - t;
    return ((l & 7) << 3) | (l >> 3);
}

__device__ __forceinline__ float silu_f(float x)     { return x / (1.0f + __expf(-x)); }
__device__ __forceinline__ float softplus_f(float x) { return (x > 20.0f) ? x : logf(1.0f + __expf(x)); }

__device__ __forceinline__ v8f wmma4(v2f a, v2f b, v8f c) {
    // D = A(16x4 f32) * B(4x16 f32) + C, full fp32
    return __builtin_amdgcn_wmma_f32_16x16x4_f32(false, a, false, b, (short)0, c, false, false);
}

extern "C" __global__ void __launch_bounds__(256)
w2dmamba_kernel(const float* __restrict__ x,     const float* __restrict__ ipw,
                const float* __restrict__ convw, const float* __restrict__ convb,
                const float* __restrict__ xpw,   const float* __restrict__ dtw,
                const float* __restrict__ dtb,   const float* __restrict__ alogs,
                const float* __restrict__ dvec,  const float* __restrict__ lng,
                const float* __restrict__ lnb,   const float* __restrict__ outw,
                float* __restrict__ out)
{
    extern __shared__ float sm[];
    const int tid  = threadIdx.x;
    const int wave = tid >> 5;
    const int lane = tid & 31;
    const int ln   = lane & 15;            // row/col within 16
    const int kb   = (lane >> 4) << 1;     // K base: 0 or 2 (A/B operand layout)
    const int mro  = (lane >> 4) << 3;     // D-row offset: 0 or 8

    const int wid = blockIdx.x;
    const int bb  = wid >> 8;              // batch
    const int whh = (wid >> 4) & 15;       // window row
    const int www = wid & 15;              // window col

    // ---- phase 0: load x window -> x_lm[l][m] ----
    for (int t = 0; t < 16; ++t) {
        int idx = t * 256 + tid;
        int m = idx >> 6, l = idx & 63;
        int h = whh * 8 + (l >> 3), w = www * 8 + (l & 7);
        sm[S_XLM + l * 66 + m] = x[((bb * DM + m) * 128 + h) * 128 + w];
    }
    __syncthreads();

    // ---- phase 1: in_proj  xz[l][c] = sum_m x[l][m]*ipw[c][m]  (M=64,N=256,K=64)
    // wave owns one M-row (A fragments loaded once), 8 N-tiles, B double-buffered.
    {
        const int mt   = wave & 3;
        const int nt0  = (wave >> 2) * 8;
        const int lrow = mt * 16 + ln;
        v2f areg[16];
        #pragma unroll
        for (int ks = 0; ks < 16; ++ks)
            areg[ks] = *(const v2f*)&sm[S_XLM + lrow * 66 + ks * 4 + kb];
        v2f breg[2][16];
        #pragma unroll
        for (int ks = 0; ks < 16; ++ks)
            breg[0][ks] = *(const v2f*)&ipw[(nt0 * 16 + ln) * DM + ks * 4 + kb];
        for (int i = 0; i < 8; ++i) {
            const int cur = i & 1;
            if (i < 7) {
                #pragma unroll
                for (int ks = 0; ks < 16; ++ks)
                    breg[cur ^ 1][ks] = *(const v2f*)&ipw[((nt0 + i + 1) * 16 + ln) * DM + ks * 4 + kb];
            }
            v8f acc = {};
            #pragma unroll
            for (int ks = 0; ks < 16; ++ks)
                acc = wmma4(areg[ks], breg[cur][ks], acc);
            const int ccol = (nt0 + i) * 16 + ln;
            #pragma unroll
            for (int j = 0; j < 8; ++j) {
                int l = mt * 16 + j + mro;
                if (ccol < DI) sm[S_XC + ccol * 66 + l] = acc[j];          // xc [c][l]
                else           sm[S_Z + l * 130 + (ccol - DI)] = acc[j];   // z  [l][c]
            }
        }
    }
    __syncthreads();

    // ---- phase 2: depthwise 3x3 conv + bias + SiLU -> xcv[c][l] ----
    {
        const int c = tid >> 1;
        const int lbase = (tid & 1) * 32;
        float wreg[9];
        #pragma unroll
        for (int n = 0; n < 9; ++n) wreg[n] = convw[c * 9 + n];
        const float bias = convb[c];
        for (int l0 = 0; l0 < 32; ++l0) {
            int l = lbase + l0;
            int i = l >> 3, j = l & 7;
            float acc = bias;
            #pragma unroll
            for (int di = 0; di < 3; ++di)
                #pragma unroll
                for (int dj = 0; dj < 3; ++dj) {
                    int ii = i + di - 1, jj = j + dj - 1;
                    if (ii >= 0 && ii < 8 && jj >= 0 && jj < 8)
                        acc += sm[S_XC + c * 66 + ii * 8 + jj] * wreg[di * 3 + dj];
                }
            sm[S_XCV + c * 66 + l] = silu_f(acc);
        }
    }
    __syncthreads();
    // zero ymerge (overlaps xc, now dead)
    for (int idx = tid; idx < 8448; idx += 256) sm[S_YM + idx] = 0.0f;
    __syncthreads();

    const v2f bz = {0.0f, 0.0f};

    // ---- phase 3: per-direction x_proj -> dt-proj -> selective scan ----
    for (int k = 0; k < 4; ++k) {
        // x_proj: xdblT[l][c] = sum_d xs[k][d][l]*xpw[k][c][d]  (M=64,N=48pad,K=128)
        for (int tile = wave; tile < 12; tile += 8) {
            const int mt = tile / 3, nt = tile % 3;
            const int lrow = mt * 16 + ln;
            const int pm   = posmap(k, lrow);
            const int cc   = nt * 16 + ln;
            const bool bv  = cc < (DR + 2 * NDS);
            const float* bgp = xpw + (k * 36 + cc) * DI;
            v2f ar[2][8], br[2][8];
            #pragma unroll
            for (int q = 0; q < 8; ++q) {
                int d0 = q * 4 + kb;
                ar[0][q].x = sm[S_XCV + d0 * 66 + pm];
                ar[0][q].y = sm[S_XCV + (d0 + 1) * 66 + pm];
                br[0][q]   = bv ? *(const v2f*)&bgp[d0] : bz;
            }
            v8f acc = {};
            for (int ch = 0; ch < 4; ++ch) {
                const int cur = ch & 1;
                if (ch < 3) {
                    #pragma unroll
                    for (int q = 0; q < 8; ++q) {
                        int d0 = (ch + 1) * 32 + q * 4 + kb;
                        ar[cur ^ 1][q].x = sm[S_XCV + d0 * 66 + pm];
                        ar[cur ^ 1][q].y = sm[S_XCV + (d0 + 1) * 66 + pm];
                        br[cur ^ 1][q]   = bv ? *(const v2f*)&bgp[d0] : bz;
                    }
                }
                #pragma unroll
                for (int q = 0; q < 8; ++q)
                    acc = wmma4(ar[cur][q], br[cur][q], acc);
            }
            #pragma unroll
            for (int j = 0; j < 8; ++j) {
                int l = mt * 16 + j + mro;
                sm[S_XDBL + l * 48 + cc] = acc[j];
            }
        }
        __syncthreads();

        // dt-proj: dt[l][d] = sum_r xdblT[l][r]*dtw[k][d][r]  (K=DR=4 -> ONE wmma/tile)
        for (int tile = wave; tile < 32; tile += 8) {
            const int mt = tile >> 3, nt = tile & 7;
            const int lrow = mt * 16 + ln;
            const int dch  = nt * 16 + ln;
            v2f a = *(const v2f*)&sm[S_XDBL + lrow * 48 + kb];
            v2f b = *(const v2f*)&dtw[(k * DI + dch) * DR + kb];
            v8f acc = {};
            acc = wmma4(a, b, acc);
            const float bias = dtb[k * DI + dch];
            #pragma unroll
            for (int j = 0; j < 8; ++j) {
                int l = mt * 16 + j + mro;
                sm[S_DT + dch * 66 + l] = softplus_f(acc[j] + bias);
            }
        }
        __syncthreads();

        // selective scan: one thread per channel d; 16 states in registers
        if (tid < DI) {
            const int d = tid;
            float As[NDS], hst[NDS];
            #pragma unroll
            for (int s = 0; s < NDS; ++s) {
                As[s]  = -__expf(alogs[(k * DI + d) * NDS + s]);
                hst[s] = 0.0f;
            }
            const float dsk = dvec[k * DI + d];
            for (int t = 0; t < L; ++t) {
                const int pm  = posmap(k, t);
                const float u   = sm[S_XCV + d * 66 + pm];
                const float dtv = sm[S_DT + d * 66 + t];
                const float du  = dtv * u;
                const float* rp = &sm[S_XDBL + t * 48];
                // B/C rows as 16B-aligned LDS broadcasts
                float4 b0 = *(const float4*)(rp + 4);
                float4 b1 = *(const float4*)(rp + 8);
                float4 b2 = *(const float4*)(rp + 12);
                float4 b3 = *(const float4*)(rp + 16);
                float4 c0 = *(const float4*)(rp + 20);
                float4 c1 = *(const float4*)(rp + 24);
                float4 c2 = *(const float4*)(rp + 28);
                float4 c3 = *(const float4*)(rp + 32);
                float bs[NDS] = {b0.x,b0.y,b0.z,b0.w, b1.x,b1.y,b1.z,b1.w,
                                 b2.x,b2.y,b2.z,b2.w, b3.x,b3.y,b3.z,b3.w};
                float cs[NDS] = {c0.x,c0.y,c0.z,c0.w, c1.x,c1.y,c1.z,c1.w,
                                 c2.x,c2.y,c2.z,c2.w, c3.x,c3.y,c3.z,c3.w};
                float y = 0.0f;
                #pragma unroll
                for (int s = 0; s < NDS; ++s) {
                    hst[s] = hst[s] * __expf(dtv * As[s]) + du * bs[s];
                    y += hst[s] * cs[s];
                }
                y += dsk * u;
                sm[S_YM + d * 66 + pm] += y;   // one thread per d; directions sequential
            }
        }
        __syncthreads();
    }

    // ---- phase 4: LayerNorm over DI + SiLU(z) gate -> ya[l][d] ----
    if (tid < L) {
        const int pos = tid;
        float mu = 0.0f;
        for (int d = 0; d < DI; ++d) mu += sm[S_YM + d * 66 + pos];
        mu *= (1.0f / DI);
        float var = 0.0f;
        for (int d = 0; d < DI; ++d) { float t = sm[S_YM + d * 66 + pos] - mu; var += t * t; }
        var *= (1.0f / DI);
        const float rs = rsqrtf(var + 1e-5f);
        for (int d = 0; d < DI; ++d) {
            float v  = (sm[S_YM + d * 66 + pos] - mu) * rs * lng[d] + lnb[d];
            float zv = sm[S_Z + pos * 130 + d];
            sm[S_YA + pos * 130 + d] = v * silu_f(zv);
        }
    }
    __syncthreads();

    // ---- phase 5: out_proj  out[l][m] = sum_d ya[l][d]*outw[m][d]  (M=64,N=64,K=128)
    for (int tile = wave; tile < 16; tile += 8) {
        const int mt = tile >> 2, nt = tile & 3;
        const int lrow = mt * 16 + ln;
        const int mm   = nt * 16 + ln;
        const float* agp = &sm[S_YA + lrow * 130];
        const float* bgp = outw + mm * DI;
        v2f ar[2][8], br[2][8];
        #pragma unroll
        for (int q = 0; q < 8; ++q) {
            int d0 = q * 4 + kb;
            ar[0][q] = *(const v2f*)&agp[d0];
            br[0][q] = *(const v2f*)&bgp[d0];
        }
        v8f acc = {};
        for (int ch = 0; ch < 4; ++ch) {
            const int cur = ch & 1;
            if (ch < 3) {
                #pragma unroll
                for (int q = 0; q < 8; ++q) {
                    int d0 = (ch + 1) * 32 + q * 4 + kb;
                    ar[cur ^ 1][q] = *(const v2f*)&agp[d0];
                    br[cur ^ 1][q] = *(const v2f*)&bgp[d0];
                }
            }
            #pragma unroll
            for (int q = 0; q < 8; ++q)
                acc = wmma4(ar[cur][q], br[cur][q], acc);
        }
        #pragma unroll
        for (int j = 0; j < 8; ++j) {
            int l = mt * 16 + j + mro;
            int h = whh * 8 + (l >> 3), w = www * 8 + (l & 7);
            out[((bb * DM + mm) * 128 + h) * 128 + w] = acc[j];
        }
    }
}

extern "C" void kernel_launch(void* const* d_in, const int* in_sizes, int n_in,
                              void* d_out, int out_size, void* d_ws, size_t ws_size,
                              hipStream_t stream) {
    const float* x     = (const float*)d_in[0];
    const float* ipw   = (const float*)d_in[1];
    const float* convw = (const float*)d_in[2];
    const float* convb = (const float*)d_in[3];
    const float* xpw   = (const float*)d_in[4];
    const float* dtw   = (const float*)d_in[5];
    const float* dtb   = (const float*)d_in[6];
    const float* alogs = (const float*)d_in[7];
    const float* dvec  = (const float*)d_in[8];
    const float* lng   = (const float*)d_in[9];
    const float* lnb   = (const float*)d_in[10];
    const float* outw  = (const float*)d_in[11];
    float* out = (float*)d_out;

    int B = in_sizes[0] / (DM * 128 * 128);   // 2
    int windows = B * 16 * 16;                // one workgroup per 8x8 window
    size_t smem = SMEM_FLOATS * sizeof(float);

    w2dmamba_kernel<<<dim3(windows), dim3(256), smem, stream>>>(
        x, ipw, convw, convb, xpw, dtw, dtb, alogs, dvec, lng, lnb, outw, out);
}